// GAT_21912923144583
// MI455X (gfx1250) — compile-verified
//
#include <hip/hip_runtime.h>
#include <hip/hip_bf16.h>
#include <cstdint>

// ---------------------------------------------------------------------------
// GAT 2-layer forward for MI455X (gfx1250, wave32).
// GEMMs via V_WMMA_F32_16X16X4_F32; edge softmax/aggregation via L2-resident
// float atomics (entire ~166MB working set fits the 192MB global L2).
// ---------------------------------------------------------------------------

#define NNODES 50000
#define NEDGES 800000
#define IN_F   128
#define HID_F  32
#define HEADS  8
#define OUT_F  16
#define NEG_SLOPE 0.2f

typedef float v2f __attribute__((ext_vector_type(2)));
typedef float v8f __attribute__((ext_vector_type(8)));

// ---------------- fp32 WMMA GEMM: C[M,NC] = A[M,K] @ B[K,NC] ----------------
// Row-major everywhere. M % 16 == 0, NC % 64 == 0, K % 4 == 0.
// One wave computes a 16x64 strip: 4 accumulator tiles, A fragment shared.
__global__ __launch_bounds__(256)
void wmma_gemm_f32(const float* __restrict__ A,
                   const float* __restrict__ B,
                   float* __restrict__ C,
                   int M, int K, int NC) {
    const int wavesPerBlock = blockDim.x >> 5;
    const int wave = blockIdx.x * wavesPerBlock + (threadIdx.x >> 5);
    const int lane = threadIdx.x & 31;
    const int colGroups = NC >> 6;                 // 64-wide column groups
    const int rowTile = wave / colGroups;          // wave-uniform
    const int cg      = wave % colGroups;
    if (rowTile * 16 >= M) return;                 // wave-uniform: EXEC stays all-1s

    // A fragment (16x4 f32): lanes 0-15 -> M, vgpr j holds K = 2*(lane>>4)+j
    const int m     = lane & 15;
    const int khalf = lane >> 4;
    const long arow = (long)(rowTile * 16 + m) * K;

    // B fragment (4x16 f32): lanes hold N; vgpr j holds K = 2*(lane>>4)+j
    const int n    = lane & 15;
    const int colb = cg * 64 + n;

    v8f acc0 = {}, acc1 = {}, acc2 = {}, acc3 = {};
    for (int k = 0; k < K; k += 4) {
        const int ka = k + khalf * 2;
        v2f a;
        a.x = A[arow + ka];
        a.y = A[arow + ka + 1];
        const float* __restrict__ B0 = B + (long)ka * NC + colb;
        const float* __restrict__ B1 = B0 + NC;
        v2f b0, b1, b2, b3;
        b0.x = B0[0];  b0.y = B1[0];
        b1.x = B0[16]; b1.y = B1[16];
        b2.x = B0[32]; b2.y = B1[32];
        b3.x = B0[48]; b3.y = B1[48];
        acc0 = __builtin_amdgcn_wmma_f32_16x16x4_f32(false, a, false, b0, (short)0, acc0, false, false);
        acc1 = __builtin_amdgcn_wmma_f32_16x16x4_f32(false, a, false, b1, (short)0, acc1, false, false);
        acc2 = __builtin_amdgcn_wmma_f32_16x16x4_f32(false, a, false, b2, (short)0, acc2, false, false);
        acc3 = __builtin_amdgcn_wmma_f32_16x16x4_f32(false, a, false, b3, (short)0, acc3, false, false);
    }

    // C fragment: vgpr j -> row M = j + 8*(lane>>4), col = lane&15
    const int mo = khalf * 8;
    for (int j = 0; j < 8; ++j) {
        float* __restrict__ Crow = C + (long)(rowTile * 16 + mo + j) * NC + cg * 64 + n;
        Crow[0]  = acc0[j];
        Crow[16] = acc1[j];
        Crow[32] = acc2[j];
        Crow[48] = acc3[j];
    }
}

// ---------------- helpers ----------------
__device__ __forceinline__ void atomicMaxFloat(float* addr, float val) {
    // Bit-order trick: int compare == float compare for >=0, reversed-unsigned for <0.
    if (val >= 0.0f)
        atomicMax((int*)addr, __float_as_int(val));
    else
        atomicMin((unsigned int*)addr, (unsigned int)__float_as_uint(val));
}

__global__ void fill_kernel(float* __restrict__ p, float v, int n) {
    int i = blockIdx.x * blockDim.x + threadIdx.x;
    if (i < n) p[i] = v;
}

// el[n,h] = dot(feat[n,h,:], al[h,:]);  er likewise.
__global__ void attn_coeff_kernel(const float* __restrict__ feat,
                                  const float* __restrict__ al,
                                  const float* __restrict__ ar,
                                  float* __restrict__ el,
                                  float* __restrict__ er,
                                  int N, int H, int D) {
    int i = blockIdx.x * blockDim.x + threadIdx.x;   // n*H + h
    if (i >= N * H) return;
    int n = i / H, h = i % H;
    const float* f   = feat + (long)n * H * D + (long)h * D;
    const float* alh = al + h * D;
    const float* arh = ar + h * D;
    float sl = 0.f, sr = 0.f;
    for (int d = 0; d < D; ++d) { float v = f[d]; sl += v * alh[d]; sr += v * arh[d]; }
    el[i] = sl;
    er[i] = sr;
}

__device__ __forceinline__ float edge_logit(const float* el, const float* er,
                                            int s, int d, int h, int H) {
    float v = el[s * H + h] + er[d * H + h];
    return v > 0.f ? v : NEG_SLOPE * v;
}

__global__ void edge_max_kernel(const int* __restrict__ src, const int* __restrict__ dst,
                                const float* __restrict__ el, const float* __restrict__ er,
                                float* __restrict__ emax, int E, int H) {
    int i = blockIdx.x * blockDim.x + threadIdx.x;   // e*H + h
    if (i >= E * H) return;
    int e = i / H, h = i % H;
    int s = src[e], d = dst[e];
    atomicMaxFloat(&emax[d * H + h], edge_logit(el, er, s, d, h, H));
}

__global__ void edge_sum_kernel(const int* __restrict__ src, const int* __restrict__ dst,
                                const float* __restrict__ el, const float* __restrict__ er,
                                const float* __restrict__ emax,
                                float* __restrict__ esum, int E, int H) {
    int i = blockIdx.x * blockDim.x + threadIdx.x;
    if (i >= E * H) return;
    int e = i / H, h = i % H;
    int s = src[e], d = dst[e];
    float v = edge_logit(el, er, s, d, h, H);
    atomicAdd(&esum[d * H + h], __expf(v - emax[d * H + h]));
}

__global__ void edge_aggregate_kernel(const int* __restrict__ src, const int* __restrict__ dst,
                                      const float* __restrict__ el, const float* __restrict__ er,
                                      const float* __restrict__ emax, const float* __restrict__ esum,
                                      const float* __restrict__ feat,
                                      float* __restrict__ out, int E, int H, int D) {
    int i = blockIdx.x * blockDim.x + threadIdx.x;   // e*H + h
    if (i >= E * H) return;
    int e = i / H, h = i % H;
    int s = src[e], d = dst[e];
    const float* __restrict__ f = feat + (long)s * H * D + (long)h * D;
    __builtin_prefetch(f, 0, 0);                     // global_prefetch_b8: pull gathered row early
    float v = edge_logit(el, er, s, d, h, H);
    float a = __expf(v - emax[d * H + h]) / esum[d * H + h];
    float* __restrict__ o = out + (long)d * H * D + (long)h * D;
    for (int dd = 0; dd < D; ++dd) atomicAdd(&o[dd], f[dd] * a);
}

// h = elu(h + b)  (in place), C = H*D
__global__ void elu_bias_kernel(float* __restrict__ h, const float* __restrict__ b, int N, int C) {
    int i = blockIdx.x * blockDim.x + threadIdx.x;
    if (i >= N * C) return;
    float v = h[i] + b[i % C];
    h[i] = v > 0.f ? v : (__expf(v) - 1.0f);
}

// out[n,o] = mean over heads of (agg[n,h,o] + b[h,o])
__global__ void head_mean_kernel(const float* __restrict__ agg, const float* __restrict__ b,
                                 float* __restrict__ out, int N, int H, int O) {
    int i = blockIdx.x * blockDim.x + threadIdx.x;   // n*O + o
    if (i >= N * O) return;
    int n = i / O, o = i % O;
    float s = 0.f;
    for (int h = 0; h < H; ++h) s += agg[(long)n * H * O + (long)h * O + o] + b[h * O + o];
    out[i] = s * (1.0f / H);
}

// ---------------------------------------------------------------------------
extern "C" void kernel_launch(void* const* d_in, const int* in_sizes, int n_in,
                              void* d_out, int out_size, void* d_ws, size_t ws_size,
                              hipStream_t stream) {
    const float* x   = (const float*)d_in[0];
    const float* W1  = (const float*)d_in[1];
    const float* al1 = (const float*)d_in[2];
    const float* ar1 = (const float*)d_in[3];
    const float* b1  = (const float*)d_in[4];
    const float* W2  = (const float*)d_in[5];
    const float* al2 = (const float*)d_in[6];
    const float* ar2 = (const float*)d_in[7];
    const float* b2  = (const float*)d_in[8];
    const int*   src = (const int*)d_in[9];
    const int*   dst = (const int*)d_in[10];
    float* out = (float*)d_out;

    const int N = NNODES, E = NEDGES, H = HEADS;
    const int C1 = HEADS * HID_F;   // 256
    const int C2 = HEADS * OUT_F;   // 128

    // workspace carving (all fp32, ~166 MB total -> resident in 192MB L2)
    float* feat1 = (float*)d_ws;                   // N*C1
    float* h1    = feat1 + (long)N * C1;           // N*C1
    float* feat2 = h1    + (long)N * C1;           // N*C2
    float* agg2  = feat2 + (long)N * C2;           // N*C2
    float* el1   = agg2  + (long)N * C2;           // N*H
    float* er1   = el1 + (long)N * H;
    float* emax1 = er1 + (long)N * H;
    float* esum1 = emax1 + (long)N * H;
    float* el2   = esum1 + (long)N * H;
    float* er2   = el2 + (long)N * H;
    float* emax2 = er2 + (long)N * H;
    float* esum2 = emax2 + (long)N * H;

    const float NEG_INF = __builtin_huge_valf() * -1.0f;
    const int TB = 256;
    auto blocks = [](long n, int tb) { return (int)((n + tb - 1) / tb); };

    // ---------------- Layer 1 ----------------
    {   // feat1 = x @ W1   (M=N, K=128, NC=256)
        int totalWaves = (N / 16) * (C1 / 64);
        wmma_gemm_f32<<<blocks((long)totalWaves * 32, TB), TB, 0, stream>>>(x, W1, feat1, N, IN_F, C1);
    }
    attn_coeff_kernel<<<blocks((long)N * H, TB), TB, 0, stream>>>(feat1, al1, ar1, el1, er1, N, H, HID_F);
    fill_kernel<<<blocks((long)N * H, TB), TB, 0, stream>>>(emax1, NEG_INF, N * H);
    fill_kernel<<<blocks((long)N * H, TB), TB, 0, stream>>>(esum1, 0.0f, N * H);
    fill_kernel<<<blocks((long)N * C1, TB), TB, 0, stream>>>(h1, 0.0f, N * C1);
    edge_max_kernel<<<blocks((long)E * H, TB), TB, 0, stream>>>(src, dst, el1, er1, emax1, E, H);
    edge_sum_kernel<<<blocks((long)E * H, TB), TB, 0, stream>>>(src, dst, el1, er1, emax1, esum1, E, H);
    edge_aggregate_kernel<<<blocks((long)E * H, TB), TB, 0, stream>>>(
        src, dst, el1, er1, emax1, esum1, feat1, h1, E, H, HID_F);
    elu_bias_kernel<<<blocks((long)N * C1, TB), TB, 0, stream>>>(h1, b1, N, C1);

    // ---------------- Layer 2 ----------------
    {   // feat2 = h1 @ W2   (M=N, K=256, NC=128)
        int totalWaves = (N / 16) * (C2 / 64);
        wmma_gemm_f32<<<blocks((long)totalWaves * 32, TB), TB, 0, stream>>>(h1, W2, feat2, N, C1, C2);
    }
    attn_coeff_kernel<<<blocks((long)N * H, TB), TB, 0, stream>>>(feat2, al2, ar2, el2, er2, N, H, OUT_F);
    fill_kernel<<<blocks((long)N * H, TB), TB, 0, stream>>>(emax2, NEG_INF, N * H);
    fill_kernel<<<blocks((long)N * H, TB), TB, 0, stream>>>(esum2, 0.0f, N * H);
    fill_kernel<<<blocks((long)N * C2, TB), TB, 0, stream>>>(agg2, 0.0f, N * C2);
    edge_max_kernel<<<blocks((long)E * H, TB), TB, 0, stream>>>(src, dst, el2, er2, emax2, E, H);
    edge_sum_kernel<<<blocks((long)E * H, TB), TB, 0, stream>>>(src, dst, el2, er2, emax2, esum2, E, H);
    edge_aggregate_kernel<<<blocks((long)E * H, TB), TB, 0, stream>>>(
        src, dst, el2, er2, emax2, esum2, feat2, agg2, E, H, OUT_F);

    head_mean_kernel<<<blocks((long)N * OUT_F, TB), TB, 0, stream>>>(agg2, b2, out, N, H, OUT_F);
}